// VAEMS_2559800508767
// MI455X (gfx1250) — compile-verified
//
#include <hip/hip_runtime.h>
#include <math.h>

// ---------------------------------------------------------------------------
// Problem constants (from reference)
// ---------------------------------------------------------------------------
#define B_   4096
#define IN_  1024
#define L0_  2048
#define L1_  1024
#define H_   128
#define NT_  512
#define EPS_ 1e-5f
// T = 0.1  ->  1/T = 10

typedef __attribute__((ext_vector_type(2))) float v2f;
typedef __attribute__((ext_vector_type(8))) float v8f;

// exact parameter types for the async-copy builtin (from hipcc diagnostic):
// param0: 'int __attribute__((vector_size(16))) __device__ *'  (addrspace(1))
typedef int i32x4 __attribute__((vector_size(16)));
typedef __attribute__((address_space(1))) i32x4 gbl_i32x4;
typedef __attribute__((address_space(3))) i32x4 lds_i32x4;

// ---------------------------------------------------------------------------
// Async global -> LDS copy (CDNA5 GLOBAL_LOAD_ASYNC_TO_LDS_B128, ASYNCcnt)
// ---------------------------------------------------------------------------
__device__ __forceinline__ void async_copy_b128(const float* gsrc, float* ldst)
{
    __builtin_amdgcn_global_load_async_to_lds_b128(
        (gbl_i32x4*)(uintptr_t)gsrc,
        (lds_i32x4*)(unsigned)(uintptr_t)ldst,
        0, 0);
}

// ---------------------------------------------------------------------------
// WMMA fp32 GEMM:  C[M,N] = A[M,K] @ W[K,N] (+bias)   (M,N mult of 128, K mult of 16)
// mode 0: C = acc + bias (bias may be null)
// mode 2: C = clamp(|acc + bias| * diag[col], 1e-10, 1e7)   (rate epilogue)
// Double-buffered LDS tiles filled by async b128 copies.
// ---------------------------------------------------------------------------
#define TILE_M 128
#define TILE_N 128
#define TILE_K 16
#define LDA_S  20    // [m][k] A tile row stride: 16B-aligned chunks + conflict-free frag reads
#define LDB_S  136   // [k][n] B tile row stride: 2*136 mod 64 = 16 -> half-waves hit disjoint banks

__global__ __launch_bounds__(256) void wmma_gemm_kernel(
    const float* __restrict__ A, const float* __restrict__ W,
    const float* __restrict__ bias, const float* __restrict__ diag,
    float* __restrict__ C, int M, int N, int K, int mode)
{
    __shared__ float As[2][TILE_M][LDA_S];
    __shared__ float Bs[2][TILE_K][LDB_S];

    const int tid  = threadIdx.x;
    const int lane = tid & 31;
    const int wave = tid >> 5;
    const int wm   = wave & 3;          // 4 row groups of 32
    const int wn   = wave >> 2;         // 2 col groups of 64
    const int m0   = blockIdx.y * TILE_M;
    const int n0   = blockIdx.x * TILE_N;

    const int l15   = lane & 15;
    const int khalf = (lane >> 4) * 2;  // 0 or 2: K sub-lane offset per ISA A/B layout

    v8f acc[2][4] = {};

    // stage one 128x16 A tile + 16x128 B tile: 512 b128 chunks each pair, 4/thread
    auto stage = [&](int buf, int k0) {
        #pragma unroll
        for (int j = 0; j < 2; ++j) {
            const int c  = tid * 2 + j;
            const int m  = c >> 2;
            const int kq = (c & 3) * 4;
            async_copy_b128(A + (size_t)(m0 + m) * K + (k0 + kq), &As[buf][m][kq]);
        }
        #pragma unroll
        for (int j = 0; j < 2; ++j) {
            const int c  = tid * 2 + j;
            const int kr = c >> 5;
            const int nq = (c & 31) * 4;
            async_copy_b128(W + (size_t)(k0 + kr) * N + (n0 + nq), &Bs[buf][kr][nq]);
        }
    };

    const int ntiles = K / TILE_K;
    stage(0, 0);

    for (int t = 0; t < ntiles; ++t) {
        const int buf = t & 1;
        if (t + 1 < ntiles) {
            stage(buf ^ 1, (t + 1) * TILE_K);
            __builtin_amdgcn_s_wait_asynccnt(4);   // current tile's 4 copies done
        } else {
            __builtin_amdgcn_s_wait_asynccnt(0);
        }
        __syncthreads();

        #pragma unroll
        for (int kk = 0; kk < TILE_K; kk += 4) {
            v2f a[2], b[4];
            #pragma unroll
            for (int i = 0; i < 2; ++i) {
                const int m = wm * 32 + i * 16 + l15;
                a[i] = *(const v2f*)&As[buf][m][kk + khalf];   // contiguous (k,k+1) pair
            }
            #pragma unroll
            for (int tt = 0; tt < 4; ++tt) {
                const int n = wn * 64 + tt * 16 + l15;
                b[tt][0] = Bs[buf][kk + khalf + 0][n];
                b[tt][1] = Bs[buf][kk + khalf + 1][n];
            }
            #pragma unroll
            for (int i = 0; i < 2; ++i)
                #pragma unroll
                for (int tt = 0; tt < 4; ++tt)
                    acc[i][tt] = __builtin_amdgcn_wmma_f32_16x16x4_f32(
                        false, a[i], false, b[tt], (short)0, acc[i][tt], false, false);
        }
        __syncthreads();   // all waves done reading buf before it is refilled
    }

    // epilogue: C/D layout -> VGPR r, lanes<16: row r, lanes>=16: row r+8
    const int rbase = (lane >> 4) * 8;
    #pragma unroll
    for (int i = 0; i < 2; ++i) {
        #pragma unroll
        for (int tt = 0; tt < 4; ++tt) {
            const int col = n0 + wn * 64 + tt * 16 + l15;
            const float bv = bias ? bias[col] : 0.0f;
            const float dv = (mode == 2) ? diag[col] : 0.0f;
            #pragma unroll
            for (int r = 0; r < 8; ++r) {
                const int row = m0 + wm * 32 + i * 16 + rbase + r;
                float v = acc[i][tt][r] + bv;
                if (mode == 2) {
                    v = fabsf(v) * dv;
                    v = fminf(fmaxf(v, 1e-10f), 1e7f);
                }
                C[(size_t)row * N + col] = v;
            }
        }
    }
}

// ---------------------------------------------------------------------------
// BatchNorm (training mode) folded to scale/shift: one thread per column.
// ---------------------------------------------------------------------------
__global__ void bn_stats_kernel(const float* __restrict__ Y, int rows, int cols,
                                const float* __restrict__ g, const float* __restrict__ b,
                                float* __restrict__ scale, float* __restrict__ shift)
{
    const int c = blockIdx.x * blockDim.x + threadIdx.x;
    if (c >= cols) return;
    float s = 0.f, s2 = 0.f;
    for (int r = 0; r < rows; ++r) {
        const float v = Y[(size_t)r * cols + c];
        s += v; s2 += v * v;
    }
    const float inv  = 1.0f / (float)rows;
    const float mean = s * inv;
    const float var  = fmaxf(s2 * inv - mean * mean, 0.f);
    const float sc   = g[c] * rsqrtf(var + EPS_);
    scale[c] = sc;
    shift[c] = b[c] - mean * sc;
}

__global__ void bn_relu_kernel(float* __restrict__ Y, int total4, int cmask,
                               const float* __restrict__ scale,
                               const float* __restrict__ shift)
{
    const int idx = blockIdx.x * blockDim.x + threadIdx.x;
    if (idx >= total4) return;
    const int base = idx * 4;
    const int c = base & cmask;          // cols is a power of two, 4 | cols
    float4 y  = *(const float4*)(Y + base);
    float4 sc = *(const float4*)(scale + c);
    float4 sh = *(const float4*)(shift + c);
    y.x = fmaxf(y.x * sc.x + sh.x, 0.f);
    y.y = fmaxf(y.y * sc.y + sh.y, 0.f);
    y.z = fmaxf(y.z * sc.z + sh.z, 0.f);
    y.w = fmaxf(y.w * sc.w + sh.w, 0.f);
    *(float4*)(Y + base) = y;
}

// ---------------------------------------------------------------------------
// decode = |dec_w| / |row_sum(dec_w)| ; diag = |row_sum|
// ---------------------------------------------------------------------------
__global__ __launch_bounds__(256) void decode_kernel(const float* __restrict__ dec_w,
                                                     float* __restrict__ decode,
                                                     float* __restrict__ diag)
{
    __shared__ float red[256];
    const int row = blockIdx.x;
    const float* wrow = dec_w + (size_t)row * IN_;
    float s = 0.f;
    for (int j = threadIdx.x; j < IN_; j += 256) s += wrow[j];
    red[threadIdx.x] = s;
    __syncthreads();
    for (int off = 128; off > 0; off >>= 1) {
        if (threadIdx.x < off) red[threadIdx.x] += red[threadIdx.x + off];
        __syncthreads();
    }
    const float d = fabsf(red[0]);
    if (threadIdx.x == 0) diag[row] = d;
    const float invd = 1.0f / d;
    for (int j = threadIdx.x; j < IN_; j += 256)
        decode[(size_t)row * IN_ + j] = fabsf(wrow[j]) * invd;
}

// ---------------------------------------------------------------------------
// Poisson rsample: per (b,h) sequential cumsum of exponentials, soft indicator.
// Wave-uniform early exit once sigmoid((1-times)/T) has underflowed.
// ---------------------------------------------------------------------------
__global__ __launch_bounds__(256) void poisson_kernel(const float* __restrict__ u,
                                                      const float* __restrict__ rate,
                                                      float* __restrict__ samples)
{
    const int b = blockIdx.x * 2 + (threadIdx.x >> 7);
    const int h = threadIdx.x & 127;
    const float r = rate[(size_t)b * H_ + h];
    const float inv_rate = 1.0f / r;
    const float* up = u + ((size_t)b * NT_) * H_ + h;

    float times = 0.f, s = 0.f;
    for (int t = 0; t < NT_; ++t) {
        const float uu = up[(size_t)t * H_];
        times += (-__logf(uu)) * inv_rate;
        // sigmoid((1-times)/T) with T=0.1 -> 1/(1+exp((times-1)*10))
        const float z = (times - 1.0f) * 10.0f;
        s += __builtin_amdgcn_rcpf(1.0f + __expf(z));
        if ((t & 31) == 31) {
            // after times > 4.6, contribution < sigmoid(-36) ~ 2e-16
            if (__builtin_amdgcn_ballot_w32(times < 4.6f) == 0u) break;
        }
    }
    samples[(size_t)b * H_ + h] = s;
}

// ---------------------------------------------------------------------------
// Launch pipeline
// ---------------------------------------------------------------------------
extern "C" void kernel_launch(void* const* d_in, const int* in_sizes, int n_in,
                              void* d_out, int out_size, void* d_ws, size_t ws_size,
                              hipStream_t stream)
{
    const float* x        = (const float*)d_in[0];
    const float* u        = (const float*)d_in[1];
    const float* enc1_w   = (const float*)d_in[2];
    const float* enc1_b   = (const float*)d_in[3];
    const float* bn1_g    = (const float*)d_in[4];
    const float* bn1_b    = (const float*)d_in[5];
    const float* enc2_w   = (const float*)d_in[6];
    const float* enc2_b   = (const float*)d_in[7];
    const float* bn2_g    = (const float*)d_in[8];
    const float* bn2_b    = (const float*)d_in[9];
    const float* enc3_w   = (const float*)d_in[10];
    const float* enc3_b   = (const float*)d_in[11];
    const float* bn3_g    = (const float*)d_in[12];
    const float* bn3_b    = (const float*)d_in[13];
    const float* lamb_w   = (const float*)d_in[14];
    const float* lamb_b   = (const float*)d_in[15];
    const float* dec_w    = (const float*)d_in[16];

    // outputs: (x_rec[B,IN], samples[B,H], rate[B,H], decode[H,IN]) flat
    float* out      = (float*)d_out;
    float* xrec     = out;                               // 4096*1024
    float* samples  = xrec + (size_t)B_ * IN_;           // 4096*128
    float* rate     = samples + (size_t)B_ * H_;         // 4096*128
    float* decode   = rate + (size_t)B_ * H_;            // 128*1024

    // workspace
    float* ws    = (float*)d_ws;
    float* Y1    = ws;                                   // [B,L0]
    float* Y2    = Y1 + (size_t)B_ * L0_;                // [B,L1]
    float* Y3    = Y2 + (size_t)B_ * L1_;                // [B,H]
    float* scale = Y3 + (size_t)B_ * H_;                 // [2048]
    float* shift = scale + L0_;                          // [2048]
    float* diag  = shift + L0_;                          // [128]

    const dim3 blk(256);

    // ---- layer 1 ----
    wmma_gemm_kernel<<<dim3(L0_/TILE_N, B_/TILE_M), blk, 0, stream>>>(
        x, enc1_w, enc1_b, nullptr, Y1, B_, L0_, IN_, 0);
    bn_stats_kernel<<<dim3((L0_+255)/256), blk, 0, stream>>>(Y1, B_, L0_, bn1_g, bn1_b, scale, shift);
    bn_relu_kernel<<<dim3((B_*L0_/4)/256), blk, 0, stream>>>(Y1, B_*L0_/4, L0_-1, scale, shift);

    // ---- layer 2 ----
    wmma_gemm_kernel<<<dim3(L1_/TILE_N, B_/TILE_M), blk, 0, stream>>>(
        Y1, enc2_w, enc2_b, nullptr, Y2, B_, L1_, L0_, 0);
    bn_stats_kernel<<<dim3((L1_+255)/256), blk, 0, stream>>>(Y2, B_, L1_, bn2_g, bn2_b, scale, shift);
    bn_relu_kernel<<<dim3((B_*L1_/4)/256), blk, 0, stream>>>(Y2, B_*L1_/4, L1_-1, scale, shift);

    // ---- layer 3 ----
    wmma_gemm_kernel<<<dim3(H_/TILE_N, B_/TILE_M), blk, 0, stream>>>(
        Y2, enc3_w, enc3_b, nullptr, Y3, B_, H_, L1_, 0);
    bn_stats_kernel<<<dim3(1), blk, 0, stream>>>(Y3, B_, H_, bn3_g, bn3_b, scale, shift);
    bn_relu_kernel<<<dim3((B_*H_/4)/256), blk, 0, stream>>>(Y3, B_*H_/4, H_-1, scale, shift);

    // ---- decoder normalization (produces decode output + diag) ----
    decode_kernel<<<dim3(H_), blk, 0, stream>>>(dec_w, decode, diag);

    // ---- lamb GEMM fused with rate epilogue: rate = clip(|H3@W+b| * diag, ...) ----
    wmma_gemm_kernel<<<dim3(H_/TILE_N, B_/TILE_M), blk, 0, stream>>>(
        Y3, lamb_w, lamb_b, diag, rate, B_, H_, H_, 2);

    // ---- Poisson rsample (dominant stage: streams u, log/exp per element) ----
    poisson_kernel<<<dim3(B_/2), blk, 0, stream>>>(u, rate, samples);

    // ---- reconstruction: x_rec = samples @ decode ----
    wmma_gemm_kernel<<<dim3(IN_/TILE_N, B_/TILE_M), blk, 0, stream>>>(
        samples, decode, nullptr, nullptr, xrec, B_, IN_, H_, 0);
}